// TorchCHMMSensory_4664334483616
// MI455X (gfx1250) — compile-verified
//
#include <hip/hip_runtime.h>

// Problem constants (from the reference): fixed-size CHMM.
#define L_SEQ   8192          // sequence length
#define NSTATE  4096          // states
#define NCL     8             // clones per observation = block size (4096/512)
#define CH      64            // steps per scan chunk
#define NCH     (L_SEQ / CH)  // 128 chunks

typedef float v2f __attribute__((ext_vector_type(2)));
typedef float v8f __attribute__((ext_vector_type(8)));

// Reduce over lanes {g*8..g*8+7}; every lane of the group gets the group sum.
// All live data sits in lanes 0-7, so 3 levels replace the 5-level full reduce.
__device__ __forceinline__ float wsum8(float v) {
#pragma unroll
  for (int off = 4; off > 0; off >>= 1) v += __shfl_xor(v, off, 32);
  return v;
}

__device__ __forceinline__ void repl8(float x, float v[8]) {
#pragma unroll
  for (int i = 0; i < 8; ++i) v[i] = __shfl(x, i, 32);
}

// ---------------------------------------------------------------------------
// Phase 0: gather the 8x8 transfer blocks  M_t[i][j] = T[obs[t-1]*8+i][obs[t]*8+j]
// ---------------------------------------------------------------------------
__global__ __launch_bounds__(256) void chmm_gather(const float* __restrict__ T,
                                                   const int* __restrict__ obs,
                                                   float* __restrict__ M) {
  const int idx = blockIdx.x * 256 + threadIdx.x;
  if (idx >= L_SEQ * 64) return;
  const int t = idx >> 6;
  const int e = idx & 63;
  if (t == 0) { M[idx] = 0.f; return; }          // slot 0 unused
  const int i = e >> 3, j = e & 7;
  const int op = obs[t - 1], oc = obs[t];
  M[idx] = T[(size_t)(op * NCL + i) * NSTATE + (size_t)(oc * NCL + j)];
}

// ---------------------------------------------------------------------------
// Phase A: per-chunk 8x8 matrix chain products via V_WMMA_F32_16X16X4_F32.
// P_c = M_{tbeg} * ... * M_{tend}, rescaled by entry-sum per step (direction only).
// ---------------------------------------------------------------------------
__global__ __launch_bounds__(32) void chmm_chunkprod(const float* __restrict__ M,
                                                     float* __restrict__ P) {
  __shared__ float Ml[CH * 64];   // this chunk's matrices
  __shared__ float Pl[64];        // running product (8x8 row-major)
  const int c    = blockIdx.x;
  const int lane = threadIdx.x;
  const int tbeg = c * CH + 1;
  const int tend = ((c + 1) * CH < L_SEQ) ? (c + 1) * CH : (L_SEQ - 1);
  const int nm   = tend - tbeg + 1;

  const float* gM = M + (size_t)tbeg * 64;
  for (int i = lane; i < nm * 64; i += 32) Ml[i] = gM[i];
  Pl[lane]      = ((lane & 7) == (lane >> 3)) ? 1.f : 0.f;            // identity
  Pl[lane + 32] = (((lane + 32) & 7) == ((lane + 32) >> 3)) ? 1.f : 0.f;
  __syncthreads();

  const int m  = lane & 15;   // A row / B col
  const int kh = lane >> 4;   // K half (ISA: lanes 0-15 hold K 0,1; lanes 16-31 hold K 2,3)

  for (int k = 0; k < nm; ++k) {
    const float* Mt = &Ml[k * 64];
    v2f a0 = {0.f, 0.f}, a1 = {0.f, 0.f}, b0 = {0.f, 0.f}, b1 = {0.f, 0.f};
    if (m < 8) {
      const int kb = kh * 2;
      a0.x = Pl[m * 8 + kb];          a0.y = Pl[m * 8 + kb + 1];      // K slice 0..3
      a1.x = Pl[m * 8 + kb + 4];      a1.y = Pl[m * 8 + kb + 5];      // K slice 4..7
      b0.x = Mt[(kb + 0) * 8 + m];    b0.y = Mt[(kb + 1) * 8 + m];
      b1.x = Mt[(kb + 4) * 8 + m];    b1.y = Mt[(kb + 5) * 8 + m];
    }
    v8f d = {0.f, 0.f, 0.f, 0.f, 0.f, 0.f, 0.f, 0.f};
    d = __builtin_amdgcn_wmma_f32_16x16x4_f32(false, a0, false, b0, (short)0, d, false, false);
    d = __builtin_amdgcn_wmma_f32_16x16x4_f32(false, a1, false, b1, (short)0, d, false, false);

    float s = 0.f;
#pragma unroll
    for (int r = 0; r < 8; ++r) s += d[r];   // live region: lanes 0-7, VGPRs 0-7
    s = wsum8(s);
    const float inv = __builtin_amdgcn_rcpf(s);   // per-step rescale: direction only
    __syncthreads();
    if (lane < 8) {                           // C/D layout: VGPR r = row r, lane n = col n
#pragma unroll
      for (int r = 0; r < 8; ++r) Pl[r * 8 + lane] = d[r] * inv;
    }
    __syncthreads();
  }
  if (lane < 8) {
#pragma unroll
    for (int r = 0; r < 8; ++r) P[c * 64 + r * 8 + lane] = Pl[r * 8 + lane];
  }
}

// ---------------------------------------------------------------------------
// Phase B: boundary scans over the 128 chunk products. Two independent waves:
//   wave 0:  bnd[c+1] = normalize(bnd[c] @ P_c)         (forward)
//   wave 1:  wb[c]    = normalize(P_c @ wb[c+1])        (backward)
// ---------------------------------------------------------------------------
__global__ __launch_bounds__(64) void chmm_boundary(const float* __restrict__ P,
                                                    const float* __restrict__ Pi,
                                                    const int* __restrict__ obs,
                                                    float* __restrict__ bnd,
                                                    float* __restrict__ wb,
                                                    float* __restrict__ alpha_s,
                                                    float* __restrict__ cs) {
  __shared__ float Pl[NCH * 64];   // 32 KB
  const int tid  = threadIdx.x;
  const int wave = tid >> 5;
  const int lane = tid & 31;
  for (int i = tid; i < NCH * 64; i += 64) Pl[i] = P[i];
  __syncthreads();

  float v[8];
  if (wave == 0) {
    // t = 0: a0 = normalize(Pi_x on block of obs[0]); c0 = block sum
    const int o0 = obs[0];
    const float p  = (lane < 8) ? Pi[o0 * NCL + lane] : 0.f;
    const float c0 = wsum8(p);
    const float a0 = p / c0;
    if (lane == 0) cs[0] = c0;
    if (lane < 8) { alpha_s[lane] = a0; bnd[lane] = a0; }
    repl8(a0, v);
    for (int c = 0; c < NCH; ++c) {
      float dot = 0.f;
      if (lane < 8) {
        const float* pc = &Pl[c * 64];
#pragma unroll
        for (int i = 0; i < 8; ++i) dot += v[i] * pc[i * 8 + lane];
      }
      const float nv = dot * __builtin_amdgcn_rcpf(wsum8(dot));
      if (lane < 8) bnd[(c + 1) * 8 + lane] = nv;
      repl8(nv, v);
    }
  } else {
    if (lane < 8) wb[NCH * 8 + lane] = 1.f;    // beta_{L-1} = ones (exact)
#pragma unroll
    for (int i = 0; i < 8; ++i) v[i] = 1.f;
    for (int c = NCH - 1; c >= 0; --c) {
      float dot = 0.f;
      if (lane < 8) {
        const float* pc = &Pl[c * 64];
#pragma unroll
        for (int j = 0; j < 8; ++j) dot += pc[lane * 8 + j] * v[j];
      }
      const float nv = dot * __builtin_amdgcn_rcpf(wsum8(dot));  // scale cancels in gamma
      if (lane < 8) wb[c * 8 + lane] = nv;
      repl8(nv, v);
    }
  }
}

// ---------------------------------------------------------------------------
// Phase C (fwd): exact per-step replay inside each chunk (parallel over chunks)
// a_t = (a_{t-1} @ M_t);  c_t = sum;  a_t /= c_t   (rcp scalar cancels in gamma)
// ---------------------------------------------------------------------------
__global__ __launch_bounds__(32) void chmm_replay_fwd(const float* __restrict__ M,
                                                      const float* __restrict__ bnd,
                                                      float* __restrict__ alpha_s,
                                                      float* __restrict__ cs) {
  __shared__ float Ml[CH * 64];
  const int c = blockIdx.x, lane = threadIdx.x;
  const int tbeg = c * CH + 1;
  const int tend = ((c + 1) * CH < L_SEQ) ? (c + 1) * CH : (L_SEQ - 1);
  const int nm   = tend - tbeg + 1;
  const float* gM = M + (size_t)tbeg * 64;
  for (int i = lane; i < nm * 64; i += 32) Ml[i] = gM[i];
  __syncthreads();

  float v[8];
  { const float b = (lane < 8) ? bnd[c * 8 + lane] : 0.f; repl8(b, v); }
  for (int k = 0; k < nm; ++k) {
    const int t = tbeg + k;
    const float* mt = &Ml[k * 64];
    float dot = 0.f;
    if (lane < 8) {
#pragma unroll
      for (int i = 0; i < 8; ++i) dot += v[i] * mt[i * 8 + lane];
    }
    const float cv = wsum8(dot);
    const float nv = dot * __builtin_amdgcn_rcpf(cv);
    if (lane < 8) alpha_s[(size_t)t * 8 + lane] = nv;
    if (lane == 0) cs[t] = cv;
    repl8(nv, v);
  }
}

// ---------------------------------------------------------------------------
// Phase C (bwd): beta replay, w_t = (M_{t+1} @ w_{t+1}) / c_{t+1}
// ---------------------------------------------------------------------------
__global__ __launch_bounds__(32) void chmm_replay_bwd(const float* __restrict__ M,
                                                      const float* __restrict__ wb,
                                                      const float* __restrict__ cs,
                                                      float* __restrict__ beta_s) {
  __shared__ float Ml[CH * 64];
  __shared__ float rcl[CH];
  const int c = blockIdx.x, lane = threadIdx.x;
  const int tbeg = c * CH + 1;
  const int tend = ((c + 1) * CH < L_SEQ) ? (c + 1) * CH : (L_SEQ - 1);
  const int nm   = tend - tbeg + 1;
  const float* gM = M + (size_t)tbeg * 64;
  for (int i = lane; i < nm * 64; i += 32) Ml[i] = gM[i];
  for (int i = lane; i < nm; i += 32) rcl[i] = __builtin_amdgcn_rcpf(cs[tbeg + i]);
  __syncthreads();

  float v[8];
  { const float b = (lane < 8) ? wb[(c + 1) * 8 + lane] : 0.f; repl8(b, v); }
  if (c == NCH - 1 && lane < 8) beta_s[(size_t)(L_SEQ - 1) * 8 + lane] = 1.f;
  for (int k = nm - 1; k >= 0; --k) {
    const int t1 = tbeg + k;            // uses M_{t1}, c_{t1}; produces beta_{t1-1}
    const float* mt = &Ml[k * 64];
    float dot = 0.f;
    if (lane < 8) {
#pragma unroll
      for (int j = 0; j < 8; ++j) dot += mt[lane * 8 + j] * v[j];
    }
    const float nv = dot * rcl[k];      // per-row scalar: cancels in gamma normalize
    if (lane < 8) beta_s[(size_t)(t1 - 1) * 8 + lane] = nv;
    repl8(nv, v);
  }
}

// ---------------------------------------------------------------------------
// log-likelihood = sum_t log(c_t)  (deterministic tree reduction)
// ---------------------------------------------------------------------------
__global__ __launch_bounds__(256) void chmm_loglik(const float* __restrict__ cs,
                                                   float* __restrict__ out) {
  __shared__ float red[256];
  float acc = 0.f;
  for (int i = threadIdx.x; i < L_SEQ; i += 256) acc += logf(cs[i]);
  red[threadIdx.x] = acc;
  __syncthreads();
  for (int off = 128; off > 0; off >>= 1) {
    if (threadIdx.x < off) red[threadIdx.x] += red[threadIdx.x + off];
    __syncthreads();
  }
  if (threadIdx.x == 0) out[0] = red[0];
}

// ---------------------------------------------------------------------------
// gamma: dense [L, S] output; nonzero only on the 8-state block per row.
// Coalesced streaming (non-temporal) stores — 128 MB, the HBM-bound part.
// True division here: these are the actual output values.
// ---------------------------------------------------------------------------
__global__ __launch_bounds__(256) void chmm_gamma(const float* __restrict__ alpha_s,
                                                  const float* __restrict__ beta_s,
                                                  const int* __restrict__ obs,
                                                  float* __restrict__ out1) {
  __shared__ float gsh[8];
  const int t   = blockIdx.x;
  const int tid = threadIdx.x;
  if (tid < 8) gsh[tid] = alpha_s[(size_t)t * 8 + tid] * beta_s[(size_t)t * 8 + tid];
  __syncthreads();
  float s = 0.f;
#pragma unroll
  for (int j = 0; j < 8; ++j) s += gsh[j];
  const int base = obs[t] * NCL;
  float* row = out1 + (size_t)t * NSTATE;
#pragma unroll
  for (int q = 0; q < NSTATE / 256; ++q) {
    const int col = tid + q * 256;          // wave writes 32 consecutive floats
    const int r   = col - base;
    const float val = ((unsigned)r < 8u) ? gsh[r] / s : 0.f;
    __builtin_nontemporal_store(val, row + col);
  }
}

// ---------------------------------------------------------------------------
extern "C" void kernel_launch(void* const* d_in, const int* in_sizes, int n_in,
                              void* d_out, int out_size, void* d_ws, size_t ws_size,
                              hipStream_t stream) {
  const float* T  = (const float*)d_in[0];
  const float* Pi = (const float*)d_in[1];
  const int*  obs = (const int*)d_in[2];
  // d_in[3] = n_observations (structure hardcoded: 4096 states / 512 obs = 8 clones)

  char* p = (char*)d_ws;
  float* M       = (float*)p; p += (size_t)L_SEQ * 64 * sizeof(float);   // 2 MB
  float* P       = (float*)p; p += (size_t)NCH * 64 * sizeof(float);
  float* bnd     = (float*)p; p += (size_t)(NCH + 1) * 8 * sizeof(float);
  float* wb      = (float*)p; p += (size_t)(NCH + 1) * 8 * sizeof(float);
  float* alpha_s = (float*)p; p += (size_t)L_SEQ * 8 * sizeof(float);
  float* beta_s  = (float*)p; p += (size_t)L_SEQ * 8 * sizeof(float);
  float* cs      = (float*)p; p += (size_t)L_SEQ * sizeof(float);
  (void)in_sizes; (void)n_in; (void)out_size; (void)ws_size;

  float* out = (float*)d_out;

  chmm_gather    <<<(L_SEQ * 64 + 255) / 256, 256, 0, stream>>>(T, obs, M);
  chmm_chunkprod <<<NCH, 32, 0, stream>>>(M, P);
  chmm_boundary  <<<1, 64, 0, stream>>>(P, Pi, obs, bnd, wb, alpha_s, cs);
  chmm_replay_fwd<<<NCH, 32, 0, stream>>>(M, bnd, alpha_s, cs);
  chmm_replay_bwd<<<NCH, 32, 0, stream>>>(M, wb, cs, beta_s);
  chmm_loglik    <<<1, 256, 0, stream>>>(cs, out);
  chmm_gamma     <<<L_SEQ, 256, 0, stream>>>(alpha_s, beta_s, obs, out + 1);
}